// Neigh_Agg_49323404427460
// MI455X (gfx1250) — compile-verified
//
#include <hip/hip_runtime.h>

typedef __attribute__((ext_vector_type(2))) float v2f;
typedef __attribute__((ext_vector_type(8))) float v8f;

#define DIM 64

// ---------------------------------------------------------------------------
// Kernel 1: xt = relu(x @ W^T + b) via V_WMMA_F32_16X16X4_F32 (exact f32).
// grid.x = ceil(M/16) row tiles; block = 128 threads = 4 waves, one wave per
// 16-wide column tile (DIM = 64 = 4 tiles).
// ---------------------------------------------------------------------------
__global__ __launch_bounds__(128)
void linear_relu_wmma(const float* __restrict__ x,
                      const float* __restrict__ W,
                      const float* __restrict__ b,
                      float* __restrict__ xt, int M) {
  const int lane = threadIdx.x & 31;
  const int wave = threadIdx.x >> 5;       // column tile 0..3
  const int half = lane >> 4;              // 0 or 1
  const int l    = lane & 15;
  const int row0 = blockIdx.x * 16;
  const int col0 = wave * 16;

  // Clamp reads for a possible partial last row tile; keep EXEC all-ones
  // through the WMMA sequence (ISA requirement) and predicate only stores.
  const int arow = (row0 + l < M) ? (row0 + l) : (M - 1);
  const float* __restrict__ xrow = x + (size_t)arow * DIM;
  const float* __restrict__ wrow = W + (size_t)(col0 + l) * DIM;

  v8f c = {};
#pragma unroll
  for (int k0 = 0; k0 < DIM; k0 += 4) {
    const int k = k0 + 2 * half;           // A: lane<16 -> K={k0,k0+1}; lane>=16 -> K={k0+2,k0+3}
    v2f a, bm;
    a.x  = xrow[k];     a.y  = xrow[k + 1];      // A[m][k], row-major x
    bm.x = wrow[k];     bm.y = wrow[k + 1];      // B[k][n] = W[n][k]  (B = W^T slice)
    c = __builtin_amdgcn_wmma_f32_16x16x4_f32(false, a, false, bm,
                                              (short)0, c, false, false);
  }

  const float bias = b[col0 + l];
#pragma unroll
  for (int r = 0; r < 8; ++r) {
    const int m = row0 + r + 8 * half;     // C/D layout: VGPR r -> M = r (+8 for lanes 16..31)
    if (m < M) {
      float v = c[r] + bias;
      xt[(size_t)m * DIM + col0 + l] = v > 0.f ? v : 0.f;
    }
  }
}

// ---------------------------------------------------------------------------
// Kernel 2: zero-fill (out accumulator and row_sum)
// ---------------------------------------------------------------------------
__global__ void zero_f32(float* __restrict__ p, int n) {
  int i = blockIdx.x * blockDim.x + threadIdx.x;
  if (i < n) p[i] = 0.f;
}

// ---------------------------------------------------------------------------
// Kernel 3: row_sum[src[e]] += w[e]   (one thread per edge, HW f32 atomic)
// ---------------------------------------------------------------------------
__global__ void edge_rowsum(const int* __restrict__ src,
                            const float* __restrict__ w,
                            float* __restrict__ rowsum, int E) {
  int e = blockIdx.x * blockDim.x + threadIdx.x;
  if (e < E) unsafeAtomicAdd(&rowsum[src[e]], w[e]);
}

// ---------------------------------------------------------------------------
// Kernel 4: out[src[e]] += xt[tgt[e]] * w[e]
// One wave32 per edge; each lane handles a float2 (64 floats / 32 lanes).
// xt (25.6 MB) and out (25.6 MB) stay resident in the 192 MB L2, so both the
// gather and the atomic scatter run at L2 rates, not HBM.
// ---------------------------------------------------------------------------
__global__ __launch_bounds__(256)
void edge_scatter(const int* __restrict__ ei,     // [2, E] flat: [0..E)=src, [E..2E)=tgt
                  const float* __restrict__ ew,
                  const float* __restrict__ xt,
                  float* __restrict__ out, int E) {
  const int e    = blockIdx.x * 8 + (threadIdx.x >> 5);
  const int lane = threadIdx.x & 31;
  if (e >= E) return;
  const int   src = ei[e];
  const int   tgt = ei[E + e];
  const float w   = ew[e];
  const float2 v  = ((const float2*)(xt + (size_t)tgt * DIM))[lane];
  float* __restrict__ orow = out + (size_t)src * DIM + 2 * lane;
  unsafeAtomicAdd(orow,     v.x * w);
  unsafeAtomicAdd(orow + 1, v.y * w);
}

// ---------------------------------------------------------------------------
// Kernel 5: out[n][d] /= max(row_sum[n], 1)
// ---------------------------------------------------------------------------
__global__ void finalize_div(float* __restrict__ out,
                             const float* __restrict__ rowsum, int total) {
  int i = blockIdx.x * blockDim.x + threadIdx.x;
  if (i < total) {
    float rs = rowsum[i >> 6];             // i / DIM
    rs = rs < 1.f ? 1.f : rs;
    out[i] *= __frcp_rn(1.f) == 1.f ? (1.f / rs) : (1.f / rs);
  }
}

// ---------------------------------------------------------------------------
extern "C" void kernel_launch(void* const* d_in, const int* in_sizes, int n_in,
                              void* d_out, int out_size, void* d_ws, size_t ws_size,
                              hipStream_t stream) {
  const float* x  = (const float*)d_in[0];   // [N, 64]
  const float* W  = (const float*)d_in[1];   // [64, 64]
  const float* b  = (const float*)d_in[2];   // [64]
  const int*   ei = (const int*)d_in[3];     // [2, E]
  const float* ew = (const float*)d_in[4];   // [E]

  const int N = in_sizes[0] / DIM;
  const int E = in_sizes[4];
  const int ND = N * DIM;

  float* out    = (float*)d_out;
  float* xt     = (float*)d_ws;              // N*DIM floats
  float* rowsum = xt + (size_t)ND;           // N floats

  // 1) zero accumulators
  zero_f32<<<(ND + 255) / 256, 256, 0, stream>>>(out, ND);
  zero_f32<<<(N + 255) / 256, 256, 0, stream>>>(rowsum, N);

  // 2) xt = relu(x @ W^T + b)  (WMMA f32)
  linear_relu_wmma<<<(N + 15) / 16, 128, 0, stream>>>(x, W, b, xt, N);

  // 3) weighted degree
  edge_rowsum<<<(E + 255) / 256, 256, 0, stream>>>(ei, ew, rowsum, E);

  // 4) scatter-add messages (8 edges per 256-thread block)
  edge_scatter<<<(E + 7) / 8, 256, 0, stream>>>(ei, ew, xt, out, E);

  // 5) normalize
  finalize_div<<<(ND + 255) / 256, 256, 0, stream>>>(out, rowsum, ND);
}